// OGNN_NO_STRC_16604343566808
// MI455X (gfx1250) — compile-verified
//
#include <hip/hip_runtime.h>
#include <hip/hip_bf16.h>

#define N_NODES 50000
#define N_EDGES 800000
#define IN_C    128
#define HID     128
#define OUT_C   40
#define POWER1  8

typedef __attribute__((ext_vector_type(2))) float v2f;
typedef __attribute__((ext_vector_type(8))) float v8f;

// ---------------- setup kernels ----------------

__global__ void init_kernel(float* deg, int* rowmin) {
    int i = blockIdx.x * blockDim.x + threadIdx.x;
    if (i < N_NODES) deg[i] = 0.0f;
    if (i == 0) *rowmin = 0x7fffffff;
}

__global__ void deg_kernel(const int* __restrict__ ei, float* deg, int* rowmin) {
    int e = blockIdx.x * blockDim.x + threadIdx.x;
    if (e >= N_EDGES) return;
    int r = ei[e];              // row
    int c = ei[N_EDGES + e];    // col
    atomicAdd(&deg[c], 1.0f);
    atomicMin(rowmin, r);
}

__global__ void dis_kernel(float* deg) {   // in-place: deg -> deg_inv_sqrt
    int i = blockIdx.x * blockDim.x + threadIdx.x;
    if (i >= N_NODES) return;
    float d = deg[i];
    deg[i] = (d > 0.0f) ? rsqrtf(fmaxf(d, 1.0f)) : 0.0f;
}

__global__ void norm_kernel(const int* __restrict__ ei, const float* __restrict__ dis,
                            float* __restrict__ norm) {
    int e = blockIdx.x * blockDim.x + threadIdx.x;
    if (e >= N_EDGES) return;
    norm[e] = dis[ei[e]] * dis[ei[N_EDGES + e]];
}

// ---------------- xX = x @ W_linX + b (f32 WMMA), also seeds hA ----------------
// One wave per 16x16 output tile. A layout (ISA 16x4 f32): lane<16 -> K={k0,k0+1},
// lane>=16 -> K={k0+2,k0+3}; M = lane&15. B mirrored with N = lane&15.

__global__ __launch_bounds__(32)
void linx_wmma_kernel(const float* __restrict__ x, const float* __restrict__ W,
                      const float* __restrict__ b, float* __restrict__ xX,
                      float* __restrict__ hA) {
    int tile = blockIdx.x;
    int mt = tile >> 3;          // HID/16 = 8 n-tiles
    int nt = tile & 7;
    int m0 = mt * 16, n0 = nt * 16;
    int lane = threadIdx.x;
    int half = lane >> 4;
    int l    = lane & 15;

    v8f acc = {};
    #pragma unroll
    for (int k0 = 0; k0 < IN_C; k0 += 4) {
        int ka = k0 + 2 * half;
        v2f a, bm;
        a.x  = x[(m0 + l) * IN_C + ka];
        a.y  = x[(m0 + l) * IN_C + ka + 1];
        bm.x = W[ka * HID + n0 + l];
        bm.y = W[(ka + 1) * HID + n0 + l];
        acc = __builtin_amdgcn_wmma_f32_16x16x4_f32(false, a, false, bm,
                                                    (short)0, acc, false, false);
    }
    float bias = b[n0 + l];
    #pragma unroll
    for (int i = 0; i < 8; ++i) {
        int row = m0 + half * 8 + i;
        float v = acc[i] + bias;
        xX[row * HID + n0 + l] = v;
        hA[row * HID + n0 + l] = v;
    }
}

// ---------------- SpMM iteration ----------------

__global__ void copy_kernel(const float4* __restrict__ src, float4* __restrict__ dst) {
    int i = blockIdx.x * blockDim.x + threadIdx.x;
    if (i < N_NODES * HID / 4) dst[i] = src[i];
}

// 32 lanes per edge, 4 channels (float4) per lane -> 128 channels
__global__ void spmm_scatter_kernel(const int* __restrict__ ei, const float* __restrict__ norm,
                                    const int* __restrict__ rowmin,
                                    const float* __restrict__ hsrc, float* hdst) {
    unsigned t = blockIdx.x * blockDim.x + threadIdx.x;
    if (t >= (unsigned)N_EDGES * 32u) return;
    int e  = t >> 5;
    int c4 = (t & 31) * 4;
    int r  = ei[e] - *rowmin;            // row - row.min()
    int c  = ei[N_EDGES + e];
    float s = norm[e];
    float4 v = *(const float4*)&hsrc[c * HID + c4];
    float* d = &hdst[r * HID + c4];
    atomicAdd(d + 0, s * v.x);
    atomicAdd(d + 1, s * v.y);
    atomicAdd(d + 2, s * v.z);
    atomicAdd(d + 3, s * v.w);
}

// ---------------- out = relu(pp0*xX + pp1*hX) @ W_pred + b_pred (f32 WMMA) ----

__global__ __launch_bounds__(32)
void pred_wmma_kernel(const float* __restrict__ xX, const float* __restrict__ hX,
                      const float* __restrict__ policy, const float* __restrict__ Wp,
                      const float* __restrict__ bp, float* __restrict__ out) {
    int tile = blockIdx.x;
    int mt = tile / 3;                   // 3 n-tiles cover 48 cols (mask >= OUT_C)
    int nt = tile - mt * 3;
    int m0 = mt * 16, n0 = nt * 16;
    int lane = threadIdx.x;
    int half = lane >> 4;
    int l    = lane & 15;

    // pp = softmax(policy[:2])
    float p0 = policy[0], p1 = policy[1];
    float mx = fmaxf(p0, p1);
    float e0 = expf(p0 - mx), e1 = expf(p1 - mx);
    float inv = 1.0f / (e0 + e1);
    float pp0 = e0 * inv, pp1 = e1 * inv;

    int  ncol  = n0 + l;
    bool colOk = ncol < OUT_C;
    int  nc    = colOk ? ncol : (OUT_C - 1);   // clamped address, no divergence
    float wmask = colOk ? 1.0f : 0.0f;

    v8f acc = {};
    #pragma unroll
    for (int k0 = 0; k0 < HID; k0 += 4) {
        int ka = k0 + 2 * half;
        int ra = (m0 + l) * HID;
        v2f a, bm;
        a.x  = fmaxf(pp0 * xX[ra + ka]     + pp1 * hX[ra + ka],     0.0f);
        a.y  = fmaxf(pp0 * xX[ra + ka + 1] + pp1 * hX[ra + ka + 1], 0.0f);
        bm.x = Wp[ka * OUT_C + nc] * wmask;
        bm.y = Wp[(ka + 1) * OUT_C + nc] * wmask;
        acc = __builtin_amdgcn_wmma_f32_16x16x4_f32(false, a, false, bm,
                                                    (short)0, acc, false, false);
    }
    if (colOk) {
        float bias = bp[ncol];
        #pragma unroll
        for (int i = 0; i < 8; ++i) {
            int row = m0 + half * 8 + i;
            out[row * OUT_C + ncol] = acc[i] + bias;
        }
    }
}

// ---------------- launcher ----------------

extern "C" void kernel_launch(void* const* d_in, const int* in_sizes, int n_in,
                              void* d_out, int out_size, void* d_ws, size_t ws_size,
                              hipStream_t stream) {
    const float* x      = (const float*)d_in[0];
    const int*   ei     = (const int*)d_in[1];    // [2,E] flat: rows then cols
    const float* W_linX = (const float*)d_in[2];
    const float* b_linX = (const float*)d_in[3];
    const float* policy = (const float*)d_in[4];
    const float* W_pred = (const float*)d_in[5];
    const float* b_pred = (const float*)d_in[6];
    float* out = (float*)d_out;

    char* ws = (char*)d_ws;
    size_t off = 0;
    auto alloc = [&](size_t bytes) -> void* {
        void* p = ws + off;
        off += (bytes + 255) & ~(size_t)255;
        return p;
    };
    float* deg    = (float*)alloc((size_t)N_NODES * 4);            // becomes deg_inv_sqrt
    float* norm   = (float*)alloc((size_t)N_EDGES * 4);
    float* xX     = (float*)alloc((size_t)N_NODES * HID * 4);
    float* hA     = (float*)alloc((size_t)N_NODES * HID * 4);
    float* hB     = (float*)alloc((size_t)N_NODES * HID * 4);
    int*   rowmin = (int*)alloc(256);

    init_kernel<<<(N_NODES + 255) / 256, 256, 0, stream>>>(deg, rowmin);
    deg_kernel<<<(N_EDGES + 255) / 256, 256, 0, stream>>>(ei, deg, rowmin);
    dis_kernel<<<(N_NODES + 255) / 256, 256, 0, stream>>>(deg);
    norm_kernel<<<(N_EDGES + 255) / 256, 256, 0, stream>>>(ei, deg, norm);

    linx_wmma_kernel<<<(N_NODES / 16) * (HID / 16), 32, 0, stream>>>(x, W_linX, b_linX, xX, hA);

    float* cur = hA;
    float* nxt = hB;
    for (int p = 0; p < POWER1; ++p) {
        copy_kernel<<<(N_NODES * HID / 4 + 255) / 256, 256, 0, stream>>>(
            (const float4*)xX, (float4*)nxt);                      // residual seed
        spmm_scatter_kernel<<<(N_EDGES * 32 + 255) / 256, 256, 0, stream>>>(
            ei, norm, rowmin, cur, nxt);
        float* t = cur; cur = nxt; nxt = t;
    }

    pred_wmma_kernel<<<(N_NODES / 16) * 3, 32, 0, stream>>>(
        xX, cur, policy, W_pred, b_pred, out);
}